// QuantumLinear_45234595561762
// MI455X (gfx1250) — compile-verified
//
#include <hip/hip_runtime.h>

typedef float v2f __attribute__((ext_vector_type(2)));
typedef float v8f __attribute__((ext_vector_type(8)));

#define NQ      4
#define DIM     16         // 2^NQ
#define C_CIRC  1024       // F / NQ
#define B_BATCH 512
#define OUTW    4096       // C * NQ == F

// ds_swizzle group-of-32: xor_mask=16, or=0, and=0x1f  -> swap 16-lane halves
__device__ __forceinline__ float swap_half(float v) {
  return __int_as_float(__builtin_amdgcn_ds_swizzle(__float_as_int(v), 0x401F));
}

// ---------------------------------------------------------------------------
// Kernel 1: per (circuit c, basis column j) simulate the BasicEntanglerLayers
// acting on (-i)^popcount(j) * e_j, producing column j of V_c.
// VR/VI layout: [c][j][k]  (j = contraction index for the GEMM, k = out state)
// ---------------------------------------------------------------------------
__global__ __launch_bounds__(256)
void qsim_build_V(const float* __restrict__ wts,
                  float* __restrict__ VR, float* __restrict__ VI) {
  int tid = blockIdx.x * blockDim.x + threadIdx.x;
  if (tid >= C_CIRC * DIM) return;
  int c = tid >> 4;
  int j = tid & 15;

  float re[16], im[16];
#pragma unroll
  for (int k = 0; k < 16; ++k) { re[k] = 0.f; im[k] = 0.f; }
  int p = __popc(j) & 3;                       // (-i)^p initial phase
  re[j] = (p == 0) ? 1.f : (p == 2) ? -1.f : 0.f;
  im[j] = (p == 1) ? -1.f : (p == 3) ? 1.f : 0.f;

#pragma unroll
  for (int l = 0; l < 2; ++l) {
    // RX(weights[c][l][w]) on wire w: [[c, -is], [-is, c]]
#pragma unroll
    for (int w = 0; w < 4; ++w) {
      float a = 0.5f * wts[(c * 2 + l) * 4 + w];
      float cw = __cosf(a), sw = __sinf(a);
      int bit = 1 << (3 - w);
#pragma unroll
      for (int k = 0; k < 16; ++k) {
        if (k & bit) continue;
        int k1 = k | bit;
        float r0 = re[k], i0 = im[k], r1 = re[k1], i1 = im[k1];
        re[k]  = cw * r0 + sw * i1;
        im[k]  = cw * i0 - sw * r1;
        re[k1] = cw * r1 + sw * i0;
        im[k1] = cw * i1 - sw * r0;
      }
    }
    // ring of CNOTs: control w -> target (w+1)%4
#pragma unroll
    for (int w = 0; w < 4; ++w) {
      int cb = 1 << (3 - w);
      int tb = 1 << (3 - ((w + 1) & 3));
#pragma unroll
      for (int k = 0; k < 16; ++k) {
        if ((k & cb) && !(k & tb)) {
          int k1 = k | tb;
          float tr = re[k]; re[k] = re[k1]; re[k1] = tr;
          float ti = im[k]; im[k] = im[k1]; im[k1] = ti;
        }
      }
    }
  }
  float* vr = VR + (c * 256 + j * 16);
  float* vi = VI + (c * 256 + j * 16);
#pragma unroll
  for (int k = 0; k < 16; ++k) { vr[k] = re[k]; vi[k] = im[k]; }
}

// ---------------------------------------------------------------------------
// Kernel 2: per wave, one (circuit c, 16-batch tile).
//   A = V (M=k, K=j), B = M^T (K=j, N=b)  ->  D[k][b] via 8x wmma_f32_16x16x4
//   P[k][b] = AR^2 + AI^2 ; k lives across VGPRs (r + 8*hi), b across lanes.
//   ev reduction over k: in-register signed butterfly (wires 1..3) +
//   one xor-16 half swap (wire 0). hi=0 lanes store float4 per batch.
// ---------------------------------------------------------------------------
__global__ __launch_bounds__(128)
void qsim_wmma_ev(const float* __restrict__ x,
                  const float* __restrict__ VR, const float* __restrict__ VI,
                  float* __restrict__ out) {
  __shared__ float VRl[256];
  __shared__ float VIl[256];
  const int c = blockIdx.x;
  for (int i = threadIdx.x; i < 256; i += 128) {
    VRl[i] = VR[c * 256 + i];
    VIl[i] = VI[c * 256 + i];
  }
  __syncthreads();

  const int lane = threadIdx.x & 31;
  const int wave = threadIdx.x >> 5;       // 0..3, each wave = one b-tile
  const int col  = lane & 15;              // N = batch within tile; A's M = k
  const int hi   = lane >> 4;              // half-wave -> K pair / k bit3
  const int b0   = (blockIdx.y * 4 + wave) * 16;
  const int b    = b0 + col;

  // per-wire cos/sin of x/2 for this lane's batch (one b128 load)
  const float4 xv = *(const float4*)(x + b * OUTW + c * 4);
  float cw[4], sw[4];
  {
    const float xa[4] = {xv.x, xv.y, xv.z, xv.w};
#pragma unroll
    for (int w = 0; w < 4; ++w) {
      float a = 0.5f * xa[w];
      cw[w] = __cosf(a);
      sw[w] = __sinf(a);
    }
  }

  v8f accR = {};
  v8f accI = {};
#pragma unroll
  for (int kk = 0; kk < 4; ++kk) {
    // A 16x4: lane holds A[M=col][K=2*hi+v] = V[k=col][j]
    // B 4x16: lane holds B[K=2*hi+v][N=col] = m_{b=col}[j]   (same j set)
    v2f ar, ai, bm;
    int j0 = kk * 4 + hi * 2;
#pragma unroll
    for (int v = 0; v < 2; ++v) {
      int j = j0 + v;
      float m = 1.f;
#pragma unroll
      for (int w = 0; w < 4; ++w) m *= ((j >> (3 - w)) & 1) ? sw[w] : cw[w];
      bm[v] = m;
      ar[v] = VRl[j * 16 + col];
      ai[v] = VIl[j * 16 + col];
    }
    accR = __builtin_amdgcn_wmma_f32_16x16x4_f32(false, ar, false, bm,
                                                 (short)0, accR, false, false);
    accI = __builtin_amdgcn_wmma_f32_16x16x4_f32(false, ai, false, bm,
                                                 (short)0, accI, false, false);
  }

  // probabilities: P[r] = prob of basis state k = r + 8*hi, batch b = col
  float P[8];
#pragma unroll
  for (int r = 0; r < 8; ++r) P[r] = accR[r] * accR[r] + accI[r] * accI[r];

  // signed butterflies over k bits 0..2 (wires 3,2,1) within registers
  float s01 = P[0] + P[1], d01 = P[0] - P[1];
  float s23 = P[2] + P[3], d23 = P[2] - P[3];
  float s45 = P[4] + P[5], d45 = P[4] - P[5];
  float s67 = P[6] + P[7], d67 = P[6] - P[7];
  float sa = s01 + s23, sb = s45 + s67;
  float E0 = sa + sb;                          // total prob of this half
  float E1 = sa - sb;                          // wire 1 (k bit2)
  float E2 = (s01 - s23) + (s45 - s67);        // wire 2 (k bit1)
  float E3 = (d01 + d23) + (d45 + d67);        // wire 3 (k bit0)

  // k bit3 lives on the half-wave split: combine via xor-16 swizzle
  float Q0 = swap_half(E0);
  float Q1 = swap_half(E1);
  float Q2 = swap_half(E2);
  float Q3 = swap_half(E3);
  float ev0 = hi ? (Q0 - E0) : (E0 - Q0);      // wire 0 (k bit3)
  float ev1 = E1 + Q1;
  float ev2 = E2 + Q2;
  float ev3 = E3 + Q3;

  if (hi == 0) {
    *(float4*)(out + b * OUTW + c * 4) = make_float4(ev0, ev1, ev2, ev3);
  }
}

extern "C" void kernel_launch(void* const* d_in, const int* in_sizes, int n_in,
                              void* d_out, int out_size, void* d_ws, size_t ws_size,
                              hipStream_t stream) {
  const float* x   = (const float*)d_in[0];   // (512, 4096) f32
  const float* wts = (const float*)d_in[1];   // (1024, 2, 4) f32
  float* out = (float*)d_out;                 // (512, 4096) f32

  float* VR = (float*)d_ws;                   // 1 MB
  float* VI = VR + C_CIRC * 256;              // 1 MB

  qsim_build_V<<<dim3((C_CIRC * DIM + 255) / 256), 256, 0, stream>>>(wts, VR, VI);
  qsim_wmma_ev<<<dim3(C_CIRC, B_BATCH / 64), 128, 0, stream>>>(x, VR, VI, out);
}